// TemporalGraphAttention_65103114273086
// MI455X (gfx1250) — compile-verified
//
#include <hip/hip_runtime.h>
#include <hip/hip_bf16.h>

// ---------------------------------------------------------------------------
// Temporal GAT (2-layer) for MI455X / gfx1250.
// Dense per-node linears use V_WMMA_F32_16X16X4_F32 (fp32-exact matrix pipe).
// Edge softmax/aggregation uses hardware f32/u32 global atomics (L2-resident).
// ---------------------------------------------------------------------------

typedef __attribute__((ext_vector_type(2))) float v2f;
typedef __attribute__((ext_vector_type(8))) float v8f;

#define IN_DIM   56
#define T_DIM    16
#define XC_DIM   72   // IN + T
#define FEAT     64   // H*C
#define NHEAD    4
#define CDIM     16
#define NEG_SLOPE 0.2f
#define LN_EPS   1e-5f

// ---------------- atomics helpers ----------------
__device__ __forceinline__ void atomAddF(float* p, float v) {
  __hip_atomic_fetch_add(p, v, __ATOMIC_RELAXED, __HIP_MEMORY_SCOPE_AGENT);
}
__device__ __forceinline__ void atomMaxU(unsigned* p, unsigned v) {
  __hip_atomic_fetch_max(p, v, __ATOMIC_RELAXED, __HIP_MEMORY_SCOPE_AGENT);
}
// monotone float <-> uint mapping (unsigned compare == float compare)
__device__ __forceinline__ unsigned fenc(float x) {
  int b = __float_as_int(x);
  return (b >= 0) ? ((unsigned)b | 0x80000000u) : ~(unsigned)b;
}
__device__ __forceinline__ float fdec(unsigned u) {
  return (u & 0x80000000u) ? __int_as_float((int)(u & 0x7FFFFFFFu))
                           : __int_as_float((int)~u);
}
__device__ __forceinline__ float lrelu(float t) {
  return (t >= 0.f) ? t : NEG_SLOPE * t;
}

// ---------------- fills ----------------
__global__ void fill_f32(float* p, float v, long long n) {
  long long i = (long long)blockIdx.x * blockDim.x + threadIdx.x;
  if (i < n) p[i] = v;
}
__global__ void fill_u32(unsigned* p, unsigned v, long long n) {
  long long i = (long long)blockIdx.x * blockDim.x + threadIdx.x;
  if (i < n) p[i] = v;
}

// ---------------- temporal encoder + concat ----------------
__global__ void temporal_concat(const float* __restrict__ x,
                                const float* __restrict__ tf,
                                const float* __restrict__ Wt,
                                const float* __restrict__ bt,
                                float* __restrict__ xc, int n) {
  long long idx = (long long)blockIdx.x * blockDim.x + threadIdx.x;
  if (idx >= (long long)n * XC_DIM) return;
  int node = (int)(idx / XC_DIM);
  int c    = (int)(idx % XC_DIM);
  float v;
  if (c < IN_DIM) {
    v = x[(long long)node * IN_DIM + c];
  } else {
    int j = c - IN_DIM;
    float t = tf[node] * Wt[j] + bt[j];
    v = t > 0.f ? t : 0.f;
  }
  xc[idx] = v;
}

// ---------------- WMMA fp32 GEMM: D[nrows,64] = A[nrows,K] @ B[K,64] --------
// Block: 128 threads (4 waves). Tile: 16 rows x 64 cols, wave w -> cols 16w..
template <int K>
__global__ __launch_bounds__(128)
void gemm_wmma_f32(const float* __restrict__ A, const float* __restrict__ B,
                   float* __restrict__ D, int nrows) {
  __shared__ float sB[K * 65];        // sB[k*65 + n], padded stride (banks)
  __shared__ float sA[16 * (K + 1)];  // sA[m*(K+1) + k]
  const int tid  = threadIdx.x;
  const int wave = tid >> 5;
  const int lane = tid & 31;
  const int row0 = blockIdx.x * 16;

  // stage B (weights) into LDS
  for (int i = tid; i < K * 64; i += 128) {
    int k = i >> 6, n = i & 63;
    sB[k * 65 + n] = B[i];
  }
  // stage A 16-row tile into LDS (clamp rows for safety)
  for (int i = tid; i < 16 * K; i += 128) {
    int m = i / K, k = i - m * K;
    int r = row0 + m; if (r >= nrows) r = nrows - 1;
    sA[m * (K + 1) + k] = A[(long long)r * K + k];
  }
  __syncthreads();

  v8f acc = {};
  const int mn    = lane & 15;          // M for A frag / N for B frag
  const int khalf = (lane >> 4) * 2;    // lanes 16-31 hold K+2,K+3
  const int ncol  = wave * 16 + mn;     // this wave's output column
  const int arow  = mn * (K + 1);

#pragma unroll
  for (int k0 = 0; k0 < K; k0 += 4) {
    v2f a, b;
    a.x = sA[arow + k0 + khalf];
    a.y = sA[arow + k0 + khalf + 1];
    b.x = sB[(k0 + khalf) * 65 + ncol];
    b.y = sB[(k0 + khalf + 1) * 65 + ncol];
    acc = __builtin_amdgcn_wmma_f32_16x16x4_f32(
        /*neg_a=*/false, a, /*neg_b=*/false, b,
        /*c_mod=*/(short)0, acc, /*reuse_a=*/false, /*reuse_b=*/false);
  }

  // D layout: VGPR r, lanes 0-15 -> M=r ; lanes 16-31 -> M=r+8 ; N=lane&15
  const int rbase = row0 + (lane >> 4) * 8;
#pragma unroll
  for (int r = 0; r < 8; ++r) {
    int row = rbase + r;
    if (row < nrows) D[(long long)row * FEAT + ncol] = acc[r];
  }
}

// ---------------- per-node attention scalars s,d ----------------
__global__ void compute_sd(const float* __restrict__ h,
                           const float* __restrict__ a_src,
                           const float* __restrict__ a_dst,
                           float* __restrict__ s, float* __restrict__ d, int n) {
  int idx = blockIdx.x * blockDim.x + threadIdx.x;
  if (idx >= n * NHEAD) return;
  int node = idx >> 2, hh = idx & 3;
  const float* hp = h + (long long)node * FEAT + hh * CDIM;
  float ss = 0.f, dd = 0.f;
#pragma unroll
  for (int c = 0; c < CDIM; ++c) {
    float t = hp[c];
    ss += t * a_src[hh * CDIM + c];
    dd += t * a_dst[hh * CDIM + c];
  }
  s[idx] = ss;
  d[idx] = dd;
}

// ---------------- edge pass 1: segment max ----------------
__global__ void edge_max(const int* __restrict__ src, const int* __restrict__ dst,
                         int ne, int n, const float* __restrict__ s,
                         const float* __restrict__ d, unsigned* __restrict__ m) {
  int e = blockIdx.x * blockDim.x + threadIdx.x;
  if (e >= ne + n) return;
  int u = (e < ne) ? src[e] : (e - ne);
  int v = (e < ne) ? dst[e] : (e - ne);
#pragma unroll
  for (int h = 0; h < NHEAD; ++h) {
    float t = lrelu(s[u * NHEAD + h] + d[v * NHEAD + h]);
    atomMaxU(m + v * NHEAD + h, fenc(t));
  }
}

// ---------------- edge pass 2: segment sum of exp ----------------
__global__ void edge_sum(const int* __restrict__ src, const int* __restrict__ dst,
                         int ne, int n, const float* __restrict__ s,
                         const float* __restrict__ d,
                         const unsigned* __restrict__ m, float* __restrict__ z) {
  int e = blockIdx.x * blockDim.x + threadIdx.x;
  if (e >= ne + n) return;
  int u = (e < ne) ? src[e] : (e - ne);
  int v = (e < ne) ? dst[e] : (e - ne);
#pragma unroll
  for (int h = 0; h < NHEAD; ++h) {
    float t = lrelu(s[u * NHEAD + h] + d[v * NHEAD + h]);
    float ex = __expf(t - fdec(m[v * NHEAD + h]));
    atomAddF(z + v * NHEAD + h, ex);
  }
}

// ---------------- edge pass 3: weighted aggregation (numerator) -------------
__global__ void edge_agg(const int* __restrict__ src, const int* __restrict__ dst,
                         int ne, int n, const float* __restrict__ s,
                         const float* __restrict__ d,
                         const unsigned* __restrict__ m,
                         const float* __restrict__ hfeat,
                         float* __restrict__ acc) {
  int e = blockIdx.x * blockDim.x + threadIdx.x;
  if (e >= ne + n) return;
  int u = (e < ne) ? src[e] : (e - ne);
  int v = (e < ne) ? dst[e] : (e - ne);
  float ex[NHEAD];
#pragma unroll
  for (int h = 0; h < NHEAD; ++h) {
    float t = lrelu(s[u * NHEAD + h] + d[v * NHEAD + h]);
    ex[h] = __expf(t - fdec(m[v * NHEAD + h]));
  }
  const float* hs = hfeat + (long long)u * FEAT;
  float* av = acc + (long long)v * FEAT;
#pragma unroll 8
  for (int c = 0; c < FEAT; ++c) {
    atomAddF(av + c, ex[c >> 4] * hs[c]);
  }
}

// ---------------- per-node finish: divide, bias, LayerNorm, ReLU ------------
__global__ void node_finish(const float* __restrict__ acc,
                            const float* __restrict__ z,
                            const float* __restrict__ bias,
                            const float* __restrict__ gamma,
                            const float* __restrict__ beta,
                            float* __restrict__ outp, int n) {
  int i = blockIdx.x * blockDim.x + threadIdx.x;
  if (i >= n) return;
  float vals[FEAT];
  float mu = 0.f;
#pragma unroll
  for (int c = 0; c < FEAT; ++c) {
    float t = acc[(long long)i * FEAT + c] / (z[i * NHEAD + (c >> 4)] + 1e-16f) + bias[c];
    vals[c] = t;
    mu += t;
  }
  mu *= (1.f / FEAT);
  float var = 0.f;
#pragma unroll
  for (int c = 0; c < FEAT; ++c) {
    float dlt = vals[c] - mu;
    var += dlt * dlt;
  }
  var *= (1.f / FEAT);
  float rs = rsqrtf(var + LN_EPS);
#pragma unroll
  for (int c = 0; c < FEAT; ++c) {
    float t = (vals[c] - mu) * rs * gamma[c] + beta[c];
    outp[(long long)i * FEAT + c] = t > 0.f ? t : 0.f;
  }
}

// ---------------- global mean pool: partial reduce ----------------
__global__ void pool_reduce(const float* __restrict__ x2, float* __restrict__ gsum,
                            int n) {
  __shared__ float part[256];
  int tid = threadIdx.x;
  long long idx = (long long)blockIdx.x * 256 + tid;
  long long stride = (long long)gridDim.x * 256;   // multiple of 64
  long long total = (long long)n * FEAT;
  float local = 0.f;
  for (long long i = idx; i < total; i += stride) local += x2[i];
  part[tid] = local;
  __syncthreads();
  if (tid < FEAT) {
    // same channel (tid % 64) across the 4 strided slots
    float ssum = part[tid] + part[tid + 64] + part[tid + 128] + part[tid + 192];
    atomAddF(gsum + tid, ssum);
  }
}

// ---------------- tiny MLP on pooled vector ----------------
__global__ void pool_mlp(const float* __restrict__ gsum,
                         const float* __restrict__ Wp1, const float* __restrict__ bp1,
                         const float* __restrict__ Wp2, const float* __restrict__ bp2,
                         float* __restrict__ out_ge, float inv_n) {
  __shared__ float ge[FEAT];
  __shared__ float hid[FEAT];
  int t = threadIdx.x;  // 64 threads
  ge[t] = gsum[t] * inv_n;
  __syncthreads();
  float a = bp1[t];
#pragma unroll
  for (int k = 0; k < FEAT; ++k) a += ge[k] * Wp1[k * FEAT + t];
  hid[t] = a > 0.f ? a : 0.f;
  __syncthreads();
  float o = bp2[t];
#pragma unroll
  for (int k = 0; k < FEAT; ++k) o += hid[k] * Wp2[k * FEAT + t];
  out_ge[t] = o;
}

// ---------------------------------------------------------------------------
extern "C" void kernel_launch(void* const* d_in, const int* in_sizes, int n_in,
                              void* d_out, int out_size, void* d_ws, size_t ws_size,
                              hipStream_t stream) {
  const float* x   = (const float*)d_in[0];
  const float* tf  = (const float*)d_in[1];
  const int*   ei  = (const int*)d_in[2];
  const float* Wt  = (const float*)d_in[3];
  const float* bt  = (const float*)d_in[4];
  const float* W1  = (const float*)d_in[5];
  const float* as1 = (const float*)d_in[6];
  const float* ad1 = (const float*)d_in[7];
  const float* b1  = (const float*)d_in[8];
  const float* g1  = (const float*)d_in[9];
  const float* be1 = (const float*)d_in[10];
  const float* W2  = (const float*)d_in[11];
  const float* as2 = (const float*)d_in[12];
  const float* ad2 = (const float*)d_in[13];
  const float* b2  = (const float*)d_in[14];
  const float* g2  = (const float*)d_in[15];
  const float* be2 = (const float*)d_in[16];
  const float* Wp1 = (const float*)d_in[17];
  const float* bp1 = (const float*)d_in[18];
  const float* Wp2 = (const float*)d_in[19];
  const float* bp2 = (const float*)d_in[20];

  const int N = in_sizes[0] / IN_DIM;      // 100000
  const int E = in_sizes[2] / 2;           // 1600000
  const int ETOT = E + N;                  // with self-loops

  float* out_x2 = (float*)d_out;
  float* out_ge = out_x2 + (long long)N * FEAT;

  // workspace layout (xc region reused as acc after GEMM1 consumes it)
  char* w = (char*)d_ws;
  float*    xc   = (float*)w;                 w += (long long)N * XC_DIM * 4;  // also acc
  float*    acc  = xc;                        // alias (xc only needed for GEMM1)
  float*    hbuf = (float*)w;                 w += (long long)N * FEAT * 4;
  float*    x1   = (float*)w;                 w += (long long)N * FEAT * 4;
  float*    sbuf = (float*)w;                 w += (long long)N * NHEAD * 4;
  float*    dbuf = (float*)w;                 w += (long long)N * NHEAD * 4;
  float*    zbuf = (float*)w;                 w += (long long)N * NHEAD * 4;
  unsigned* mbuf = (unsigned*)w;              w += (long long)N * NHEAD * 4;
  float*    gsum = (float*)w;                 w += 64 * 4;

  const int* srcp = ei;
  const int* dstp = ei + E;

  const int B256 = 256;
  int gridNC   = (int)(((long long)N * XC_DIM + B256 - 1) / B256);
  int gridTile = (N + 15) / 16;
  int gridSD   = (N * NHEAD + B256 - 1) / B256;
  int gridNH   = (int)(((long long)N * NHEAD + B256 - 1) / B256);
  int gridF    = (int)(((long long)N * FEAT + B256 - 1) / B256);
  int gridE    = (ETOT + B256 - 1) / B256;
  int gridN    = (N + 127) / 128;

  // ---------- temporal encoder + concat ----------
  temporal_concat<<<gridNC, B256, 0, stream>>>(x, tf, Wt, bt, xc, N);

  // ================= layer 1 =================
  gemm_wmma_f32<XC_DIM><<<gridTile, 128, 0, stream>>>(xc, W1, hbuf, N);
  compute_sd<<<gridSD, B256, 0, stream>>>(hbuf, as1, ad1, sbuf, dbuf, N);
  fill_u32<<<gridNH, B256, 0, stream>>>(mbuf, 0u, (long long)N * NHEAD);
  fill_f32<<<gridNH, B256, 0, stream>>>(zbuf, 0.f, (long long)N * NHEAD);
  fill_f32<<<gridF, B256, 0, stream>>>(acc, 0.f, (long long)N * FEAT);
  edge_max<<<gridE, B256, 0, stream>>>(srcp, dstp, E, N, sbuf, dbuf, mbuf);
  edge_sum<<<gridE, B256, 0, stream>>>(srcp, dstp, E, N, sbuf, dbuf, mbuf, zbuf);
  edge_agg<<<gridE, B256, 0, stream>>>(srcp, dstp, E, N, sbuf, dbuf, mbuf, hbuf, acc);
  node_finish<<<gridN, 128, 0, stream>>>(acc, zbuf, b1, g1, be1, x1, N);

  // ================= layer 2 =================
  gemm_wmma_f32<FEAT><<<gridTile, 128, 0, stream>>>(x1, W2, hbuf, N);
  compute_sd<<<gridSD, B256, 0, stream>>>(hbuf, as2, ad2, sbuf, dbuf, N);
  fill_u32<<<gridNH, B256, 0, stream>>>(mbuf, 0u, (long long)N * NHEAD);
  fill_f32<<<gridNH, B256, 0, stream>>>(zbuf, 0.f, (long long)N * NHEAD);
  fill_f32<<<gridF, B256, 0, stream>>>(acc, 0.f, (long long)N * FEAT);
  edge_max<<<gridE, B256, 0, stream>>>(srcp, dstp, E, N, sbuf, dbuf, mbuf);
  edge_sum<<<gridE, B256, 0, stream>>>(srcp, dstp, E, N, sbuf, dbuf, mbuf, zbuf);
  edge_agg<<<gridE, B256, 0, stream>>>(srcp, dstp, E, N, sbuf, dbuf, mbuf, hbuf, acc);
  node_finish<<<gridN, 128, 0, stream>>>(acc, zbuf, b2, g2, be2, out_x2, N);

  // ================= global mean pool + MLP =================
  fill_f32<<<1, 64, 0, stream>>>(gsum, 0.f, 64);
  pool_reduce<<<128, 256, 0, stream>>>(out_x2, gsum, N);
  pool_mlp<<<1, 64, 0, stream>>>(gsum, Wp1, bp1, Wp2, bp2, out_ge, 1.f / (float)N);
}